// OutlierLoss_88502096101542
// MI455X (gfx1250) — compile-verified
//
#include <hip/hip_runtime.h>
#include <hip/hip_bf16.h>
#include <math.h>

typedef float v2f __attribute__((ext_vector_type(2)));
typedef float v8f __attribute__((ext_vector_type(8)));

#define INV_TEMP 10.0f   // 1/0.1, exact
#define LP_W     0.1f
#define EPSN     1e-12f

// ---------- helpers ----------
__device__ __forceinline__ unsigned enc_ord(float f) {
    unsigned u = __float_as_uint(f);
    return (u & 0x80000000u) ? ~u : (u | 0x80000000u);   // monotonic float->uint
}
__device__ __forceinline__ float dec_ord(unsigned e) {
    return __uint_as_float((e & 0x80000000u) ? (e & 0x7fffffffu) : ~e);
}

// ws layout (32-bit slots):
// [0] f: max(||centroid||, eps)      [1] u: enc(max neg logit)
// [2] u: enc(max neg edge logit)     [3] f: sum exp(neg logits - max)
// [4] f: sum exp(neg edges - max)    [5] f: sum positive contrastive terms
// [6] f: n_pos                       [7] f: sum positive lp terms

// ---------- init: zero accumulators + centroid norm ----------
__global__ __launch_bounds__(256) void k_init(const float* __restrict__ cent,
                                              float* ws_f, unsigned* ws_u, int dim) {
    __shared__ float sm[256];
    int t = threadIdx.x;
    float v = (t < dim) ? cent[t] : 0.0f;
    sm[t] = v * v;
    __syncthreads();
    for (int s = 128; s > 0; s >>= 1) {
        if (t < s) sm[t] += sm[t + s];
        __syncthreads();
    }
    if (t == 0) {
        ws_f[0] = fmaxf(sqrtf(sm[0]), EPSN);
        ws_u[1] = enc_ord(-INFINITY);
        ws_u[2] = enc_ord(-INFINITY);
        ws_f[3] = 0.0f; ws_f[4] = 0.0f; ws_f[5] = 0.0f;
        ws_f[6] = 0.0f; ws_f[7] = 0.0f;
    }
}

// ---------- GEMV via WMMA f32 16x16x4: logits = (feat @ q) / TEMP ----------
// One wave owns 16 rows. A tile: lane(0..15)=row, K={0,1}; lane(16..31)=row, K={2,3}.
// B: same k-chunk of q broadcast to all 16 columns -> every column of D = logits.
__global__ __launch_bounds__(256) void k_gemv_logits(const float* __restrict__ feat,
                                                     const float* __restrict__ q,
                                                     float* __restrict__ logits) {
    __shared__ __align__(16) float qs[256];
    qs[threadIdx.x] = q[threadIdx.x];
    __syncthreads();
    const int wave    = threadIdx.x >> 5;
    const int lane    = threadIdx.x & 31;
    const int rowbase = (blockIdx.x << 7) + (wave << 4);
    const int row     = rowbase + (lane & 15);
    const int koff    = (lane >> 4) << 1;              // 0 or 2
    const float* arow = feat + (size_t)row * 256 + koff;
    const float* bvec = qs + koff;

    v8f c0 = {0.f,0.f,0.f,0.f,0.f,0.f,0.f,0.f};
    v8f c1 = c0;
#pragma unroll 8
    for (int k = 0; k < 256; k += 8) {
        v2f a0 = *(const v2f*)(arow + k);
        v2f b0 = *(const v2f*)(bvec + k);
        v2f a1 = *(const v2f*)(arow + k + 4);
        v2f b1 = *(const v2f*)(bvec + k + 4);
        c0 = __builtin_amdgcn_wmma_f32_16x16x4_f32(false, a0, false, b0, (short)0, c0, false, false);
        c1 = __builtin_amdgcn_wmma_f32_16x16x4_f32(false, a1, false, b1, (short)0, c1, false, false);
    }
    v8f c = (c0 + c1) * INV_TEMP;
    // D col 0: lane0 holds rows 0..7 (VGPR r = M=r), lane16 holds rows 8..15
    if (lane == 0) {
#pragma unroll
        for (int r = 0; r < 8; ++r) logits[rowbase + r] = c[r];
    } else if (lane == 16) {
#pragma unroll
        for (int r = 0; r < 8; ++r) logits[rowbase + 8 + r] = c[r];
    }
}

// ---------- scores: dot(raw_i, cent) via WMMA + self-norm via VALU ----------
__global__ __launch_bounds__(256) void k_scores(const float* __restrict__ raw,
                                                const float* __restrict__ cent,
                                                const float* __restrict__ ws_f,
                                                float* __restrict__ scores) {
    __shared__ __align__(16) float cs[256];
    __shared__ float dotbuf[8][16];
    __shared__ float selfbuf[8][32];
    cs[threadIdx.x] = cent[threadIdx.x];
    __syncthreads();
    const int wave    = threadIdx.x >> 5;
    const int lane    = threadIdx.x & 31;
    const int rowbase = (blockIdx.x << 7) + (wave << 4);
    const int row     = rowbase + (lane & 15);
    const int koff    = (lane >> 4) << 1;
    const float* arow = raw + (size_t)row * 256 + koff;
    const float* bvec = cs + koff;

    v8f c0 = {0.f,0.f,0.f,0.f,0.f,0.f,0.f,0.f};
    v8f c1 = c0;
    float selfacc = 0.0f;
#pragma unroll 8
    for (int k = 0; k < 256; k += 8) {
        v2f a0 = *(const v2f*)(arow + k);
        v2f b0 = *(const v2f*)(bvec + k);
        v2f a1 = *(const v2f*)(arow + k + 4);
        v2f b1 = *(const v2f*)(bvec + k + 4);
        selfacc += a0.x * a0.x + a0.y * a0.y + a1.x * a1.x + a1.y * a1.y;
        c0 = __builtin_amdgcn_wmma_f32_16x16x4_f32(false, a0, false, b0, (short)0, c0, false, false);
        c1 = __builtin_amdgcn_wmma_f32_16x16x4_f32(false, a1, false, b1, (short)0, c1, false, false);
    }
    v8f c = c0 + c1;
    selfbuf[wave][lane] = selfacc;
    if (lane == 0) {
#pragma unroll
        for (int r = 0; r < 8; ++r) dotbuf[wave][r] = c[r];
    } else if (lane == 16) {
#pragma unroll
        for (int r = 0; r < 8; ++r) dotbuf[wave][8 + r] = c[r];
    }
    __syncthreads();
    if (lane < 16) {
        float ns  = selfbuf[wave][lane] + selfbuf[wave][lane + 16];
        float nr  = fmaxf(sqrtf(ns), EPSN);
        float rcn = ws_f[0];
        scores[rowbase + lane] = dotbuf[wave][lane] / (nr * rcn);
    }
}

// ---------- max over entries with lab==0 (+ count of lab>0) ----------
__global__ __launch_bounds__(256) void k_negmax(const float* __restrict__ x,
                                                const int* __restrict__ lab, int n,
                                                float scale, unsigned* maxdst, float* cntdst) {
    __shared__ float sm[256];
    __shared__ float sc[256];
    int t = threadIdx.x;
    float m = -INFINITY, cnt = 0.0f;
    for (int i = blockIdx.x * blockDim.x + t; i < n; i += gridDim.x * blockDim.x) {
        float v = x[i] * scale;
        if (lab[i] == 0) m = fmaxf(m, v); else cnt += 1.0f;
    }
    sm[t] = m; sc[t] = cnt;
    __syncthreads();
    for (int s = 128; s > 0; s >>= 1) {
        if (t < s) { sm[t] = fmaxf(sm[t], sm[t + s]); sc[t] += sc[t + s]; }
        __syncthreads();
    }
    if (t == 0) {
        atomicMax(maxdst, enc_ord(sm[0]));
        if (cntdst) atomicAdd(cntdst, sc[0]);
    }
}

// ---------- sum exp(x-max) over entries with lab==0 ----------
__global__ __launch_bounds__(256) void k_negsumexp(const float* __restrict__ x,
                                                   const int* __restrict__ lab, int n,
                                                   float scale, const unsigned* maxsrc,
                                                   float* dst) {
    __shared__ float sm[256];
    int t = threadIdx.x;
    float m = dec_ord(*maxsrc);
    float s = 0.0f;
    for (int i = blockIdx.x * blockDim.x + t; i < n; i += gridDim.x * blockDim.x)
        if (lab[i] == 0) s += __expf(x[i] * scale - m);
    sm[t] = s;
    __syncthreads();
    for (int w = 128; w > 0; w >>= 1) {
        if (t < w) sm[t] += sm[t + w];
        __syncthreads();
    }
    if (t == 0) atomicAdd(dst, sm[0]);
}

// ---------- sum over positives of logaddexp(x, neg_lse) - x = log1p(exp(neg_lse - x)) ----------
// lab==nullptr means "first n entries are all positive" (lp path).
__global__ __launch_bounds__(256) void k_pos_terms(const float* __restrict__ x,
                                                   const int* __restrict__ lab, int n,
                                                   float scale, const unsigned* maxsrc,
                                                   const float* sumsrc, float* dst) {
    __shared__ float sm[256];
    int t = threadIdx.x;
    float neg_lse = dec_ord(*maxsrc) + logf(*sumsrc);
    float s = 0.0f;
    for (int i = blockIdx.x * blockDim.x + t; i < n; i += gridDim.x * blockDim.x)
        if (!lab || lab[i] > 0) s += log1pf(__expf(neg_lse - x[i] * scale));
    sm[t] = s;
    __syncthreads();
    for (int w = 128; w > 0; w >>= 1) {
        if (t < w) sm[t] += sm[t + w];
        __syncthreads();
    }
    if (t == 0) atomicAdd(dst, sm[0]);
}

// ---------- final scalars ----------
__global__ void k_final(const float* __restrict__ ws_f, float* __restrict__ out,
                        int n, float pcap) {
    float contras = ws_f[5] / ws_f[6];
    float lp      = ws_f[7] / pcap;
    out[0]     = contras + LP_W * lp;
    out[n + 1] = contras;
    out[n + 2] = lp;
}

extern "C" void kernel_launch(void* const* d_in, const int* in_sizes, int n_in,
                              void* d_out, int out_size, void* d_ws, size_t ws_size,
                              hipStream_t stream) {
    const float* feat   = (const float*)d_in[0];
    const float* q      = (const float*)d_in[1];
    const float* eprob  = (const float*)d_in[2];
    const int*   elab   = (const int*)d_in[3];
    const int*   labels = (const int*)d_in[6];
    const float* raw    = (const float*)d_in[7];
    const float* cent   = (const float*)d_in[8];

    const int N   = in_sizes[6];   // 65536
    const int E   = in_sizes[2];   // 262144
    const int DIM = in_sizes[1];   // 256
    const int PCAP = 8192;

    float*    out    = (float*)d_out;
    float*    logits = out + 1;            // scratch; overwritten by scores later
    float*    ws_f   = (float*)d_ws;
    unsigned* ws_u   = (unsigned*)d_ws;

    // 0. zero accumulators + centroid norm
    k_init<<<1, 256, 0, stream>>>(cent, ws_f, ws_u, DIM);
    // 1. logits = feat @ q / TEMP   (WMMA GEMV, streams 64 MB)
    k_gemv_logits<<<N / 128, 256, 0, stream>>>(feat, q, logits);
    // 2. contrastive LSE over negatives (+ n_pos count), then positive terms
    k_negmax<<<256, 256, 0, stream>>>(logits, labels, N, 1.0f, ws_u + 1, ws_f + 6);
    k_negsumexp<<<256, 256, 0, stream>>>(logits, labels, N, 1.0f, ws_u + 1, ws_f + 3);
    k_pos_terms<<<256, 256, 0, stream>>>(logits, labels, N, 1.0f, ws_u + 1, ws_f + 3, ws_f + 5);
    // 3. link-prediction LSE over negative edges, then positive terms (first PCAP)
    k_negmax<<<256, 256, 0, stream>>>(eprob, elab, E, INV_TEMP, ws_u + 2, (float*)nullptr);
    k_negsumexp<<<256, 256, 0, stream>>>(eprob, elab, E, INV_TEMP, ws_u + 2, ws_f + 4);
    k_pos_terms<<<64, 256, 0, stream>>>(eprob, (const int*)nullptr, PCAP, INV_TEMP,
                                        ws_u + 2, ws_f + 4, ws_f + 7);
    // 4. cosine scores (WMMA dot + VALU self-norm); overwrites logits scratch
    k_scores<<<N / 128, 256, 0, stream>>>(raw, cent, ws_f, out + 1);
    // 5. final scalars
    k_final<<<1, 1, 0, stream>>>(ws_f, out, N, (float)PCAP);
}